// DiagonalLinear_7653631722083
// MI455X (gfx1250) — compile-verified
//
#include <hip/hip_runtime.h>
#include <stdint.h>

typedef float v4f __attribute__((ext_vector_type(4)));

#define THREADS 256

// Specialized streaming kernel: d == CHUNKS * THREADS * 4 columns.
// Each thread owns CHUNKS float4 column-slots, stages w/b for those slots
// into LDS with CDNA5 async global->LDS b128 DMA, hoists them to registers,
// then streams rows with non-temporal b128 loads/stores.
template <int CHUNKS>
__global__ __launch_bounds__(THREADS) void diag_linear_kernel(
    const float* __restrict__ x, const float* __restrict__ w,
    const float* __restrict__ bias, float* __restrict__ out, int batch, int d) {
  __shared__ float lds[2 * CHUNKS * THREADS * 4];  // [w | b], 4 floats/slot

  const int t = threadIdx.x;
  const uint32_t half_bytes = (uint32_t)(CHUNKS * THREADS * 16);
  const uint32_t lds0 = (uint32_t)(uintptr_t)(void*)&lds[0];  // LDS byte offset

  // ---- CDNA5 async global -> LDS staging of w and b (ASYNCcnt path) ----
  {
    const uint64_t wbase = (uint64_t)(uintptr_t)(const void*)w;
    const uint64_t bbase = (uint64_t)(uintptr_t)(const void*)bias;
#pragma unroll
    for (int i = 0; i < CHUNKS; ++i) {
      const uint32_t byteoff = (uint32_t)((t + i * THREADS) * 16);
      const uint32_t lds_w = lds0 + byteoff;
      const uint32_t lds_b = lds0 + half_bytes + byteoff;
      asm volatile("global_load_async_to_lds_b128 %0, %1, %2"
                   :
                   : "v"(lds_w), "v"(byteoff), "s"(wbase)
                   : "memory");
      asm volatile("global_load_async_to_lds_b128 %0, %1, %2"
                   :
                   : "v"(lds_b), "v"(byteoff), "s"(bbase)
                   : "memory");
    }
#if __has_builtin(__builtin_amdgcn_s_wait_asynccnt)
    __builtin_amdgcn_s_wait_asynccnt(0);
#else
    asm volatile("s_wait_asynccnt 0" ::: "memory");
#endif
    __syncthreads();
  }

  // ---- hoist this thread's w/b slots into registers (ds_load_b128) ----
  v4f wv[CHUNKS], bv[CHUNKS];
  {
    const v4f* ldsW = (const v4f*)&lds[0];
    const v4f* ldsB = (const v4f*)&lds[CHUNKS * THREADS * 4];
#pragma unroll
    for (int i = 0; i < CHUNKS; ++i) {
      wv[i] = ldsW[t + i * THREADS];
      bv[i] = ldsB[t + i * THREADS];
    }
  }

  // ---- stream rows: NT b128 load x, fma, NT b128 store out ----
  for (int r = blockIdx.x; r < batch; r += gridDim.x) {
    const v4f* __restrict__ xr = (const v4f*)(x + (size_t)r * (size_t)d);
    v4f* __restrict__ orow = (v4f*)(out + (size_t)r * (size_t)d);
    v4f xv[CHUNKS];
#pragma unroll
    for (int i = 0; i < CHUNKS; ++i)
      xv[i] = __builtin_nontemporal_load(&xr[t + i * THREADS]);
#pragma unroll
    for (int i = 0; i < CHUNKS; ++i) {
      v4f o = xv[i] * wv[i] + bv[i];
      __builtin_nontemporal_store(o, &orow[t + i * THREADS]);
    }
  }
}

// Generic fallback for shapes the template does not cover.
__global__ void diag_linear_generic(const float* __restrict__ x,
                                    const float* __restrict__ w,
                                    const float* __restrict__ bias,
                                    float* __restrict__ out, long long n,
                                    int d) {
  long long i = (long long)blockIdx.x * blockDim.x + threadIdx.x;
  const long long stride = (long long)gridDim.x * blockDim.x;
  for (; i < n; i += stride) {
    const int c = (int)(i % (long long)d);
    out[i] = x[i] * w[c] + bias[c];
  }
}

extern "C" void kernel_launch(void* const* d_in, const int* in_sizes, int n_in,
                              void* d_out, int out_size, void* d_ws,
                              size_t ws_size, hipStream_t stream) {
  const float* x = (const float*)d_in[0];
  const float* w = (const float*)d_in[1];
  const float* b = (const float*)d_in[2];
  float* out = (float*)d_out;

  const long long n = (long long)in_sizes[0];
  const int d = in_sizes[1];
  const int batch = (int)(n / (long long)d);

  const int cols_per_block = THREADS * 4;  // 1024 columns per CHUNK
  if (d > 0 && (d % cols_per_block) == 0) {
    const int chunks = d / cols_per_block;
    int grid = batch < 4096 ? batch : 4096;
    if (grid < 1) grid = 1;
    switch (chunks) {
      case 1:
        diag_linear_kernel<1><<<grid, THREADS, 0, stream>>>(x, w, b, out, batch, d);
        return;
      case 2:
        diag_linear_kernel<2><<<grid, THREADS, 0, stream>>>(x, w, b, out, batch, d);
        return;
      case 4:  // d == 4096 (the reference shape)
        diag_linear_kernel<4><<<grid, THREADS, 0, stream>>>(x, w, b, out, batch, d);
        return;
      case 8:
        diag_linear_kernel<8><<<grid, THREADS, 0, stream>>>(x, w, b, out, batch, d);
        return;
      default:
        break;
    }
  }

  // Fallback path
  long long total = n;
  int grid = (int)((total + THREADS - 1) / THREADS);
  if (grid > 65535) grid = 65535;
  if (grid < 1) grid = 1;
  diag_linear_generic<<<grid, THREADS, 0, stream>>>(x, w, b, out, total, d);
}